// GaussianStep_46445776339644
// MI455X (gfx1250) — compile-verified
//
#include <hip/hip_runtime.h>
#include <stdint.h>

#define BIGS 36  // row stride (floats) for 32-col LDS matrices: rows 16B-aligned
                 // (needed by async B128-to-LDS), and 36*L mod 64 distinct for
                 // L=0..15 with half-wave column sets disjoint mod 4 ->
                 // conflict-free WMMA fragment reads on 64 banks.

typedef __attribute__((ext_vector_type(2))) float v2f;
typedef __attribute__((ext_vector_type(8))) float v8f;
typedef __attribute__((ext_vector_type(4))) int v4i;

typedef __attribute__((address_space(1))) v4i* gptr_b128;
typedef __attribute__((address_space(3))) v4i* lptr_b128;

#if __has_builtin(__builtin_amdgcn_global_load_async_to_lds_b128)
#define HAVE_ASYNC 1
#endif

__device__ __forceinline__ void async_b128(const float* g, float* l) {
#ifdef HAVE_ASYNC
  // generic->as(1): same 64-bit value; generic->as(3): LDS offset = low 32 bits
  __builtin_amdgcn_global_load_async_to_lds_b128(
      (gptr_b128)(uintptr_t)g, (lptr_b128)(uint32_t)(uintptr_t)l, 0, 0);
#endif
}

template <int N>
__device__ __forceinline__ void wait_async() {
#ifdef HAVE_ASYNC
  asm volatile("" ::: "memory");
#if __has_builtin(__builtin_amdgcn_s_wait_asynccnt)
  __builtin_amdgcn_s_wait_asynccnt(N);
#else
  asm volatile("s_wait_asynccnt %0" ::"i"(N) : "memory");
#endif
  asm volatile("" ::: "memory");
#endif
}

// Stage nchunks*32 float4 chunks of a row-major [rows][32] matrix into LDS with
// padded row stride BIGS.  Async path: no VGPR round trip, tracked by ASYNCcnt.
template <int NCH>
__device__ __forceinline__ void stage_mat(const float* __restrict__ g,
                                          float* __restrict__ s, int lane) {
#ifdef HAVE_ASYNC
#pragma unroll
  for (int t = 0; t < NCH; ++t) {
    int e = (t * 32 + lane) * 4;
    async_b128(g + e, s + (e >> 5) * BIGS + (e & 31));
  }
#else
  const float4* g4 = reinterpret_cast<const float4*>(g);
#pragma unroll
  for (int t = 0; t < NCH; ++t) {
    float4 v = g4[t * 32 + lane];
    int e = (t * 32 + lane) * 4;
    int r = e >> 5, c = e & 31;
    s[r * BIGS + c + 0] = v.x; s[r * BIGS + c + 1] = v.y;
    s[r * BIGS + c + 2] = v.z; s[r * BIGS + c + 3] = v.w;
  }
#endif
}

__device__ __forceinline__ v8f wmma4(v2f a, v2f b, v8f c) {
  // V_WMMA_F32_16X16X4_F32: D = A(16x4 f32) * B(4x16 f32) + C(16x16 f32)
  return __builtin_amdgcn_wmma_f32_16x16x4_f32(false, a, false, b, (short)0, c,
                                               false, false);
}

struct Acc { v8f t[2][2]; };

__device__ __forceinline__ void acc_zero(Acc& a) {
  v8f z = {0.f, 0.f, 0.f, 0.f, 0.f, 0.f, 0.f, 0.f};
  a.t[0][0] = z; a.t[0][1] = z; a.t[1][0] = z; a.t[1][1] = z;
}

// acc(TI x TJ tiles of 16x16) += A(16*TI x KD) * B(KD x 16*TJ).
// If BT, Bm is stored transposed (16*TJ rows x KD cols): transpose free via
// addressing.  Fragment lane layouts (f32 16x16x4):
//   A 16x4: lanes 0-15 rows M, v0=K,   v1=K+1 ; lanes 16-31 v0=K+2, v1=K+3
//   B 4x16: v0 = row K (lanes 0-15) / K+2 (lanes 16-31), v1 = K+1 / K+3
template <bool BT, int KD, int TI, int TJ>
__device__ __forceinline__ void mm_tiles(const float* __restrict__ A, int sa,
                                         const float* __restrict__ Bm, int sb,
                                         int lane, Acc& acc) {
  const int mrow = lane & 15;
  const int ksel = (lane >> 4) << 1;  // 0 for lanes 0-15, 2 for lanes 16-31
#pragma unroll
  for (int ti = 0; ti < TI; ++ti) {
#pragma unroll
    for (int tj = 0; tj < TJ; ++tj) {
      v8f c = acc.t[ti][tj];
#pragma unroll
      for (int k = 0; k < KD; k += 4) {
        v2f av, bv;
        av.x = A[(ti * 16 + mrow) * sa + k + ksel];
        av.y = A[(ti * 16 + mrow) * sa + k + ksel + 1];
        if (BT) {
          bv.x = Bm[(tj * 16 + mrow) * sb + k + ksel];
          bv.y = Bm[(tj * 16 + mrow) * sb + k + ksel + 1];
        } else {
          bv.x = Bm[(k + ksel) * sb + tj * 16 + mrow];
          bv.y = Bm[(k + ksel + 1) * sb + tj * 16 + mrow];
        }
        c = wmma4(av, bv, c);
      }
      acc.t[ti][tj] = c;
    }
  }
}

// C/D 16x16 layout: VGPR v -> row v (lanes 0-15, col=lane) / row v+8 (lanes 16-31)
__device__ __forceinline__ void store32_lds(const Acc& acc, float* D, int sd,
                                            int lane) {
  const int half = (lane >> 4) * 8;
  const int col = lane & 15;
#pragma unroll
  for (int ti = 0; ti < 2; ++ti)
#pragma unroll
    for (int tj = 0; tj < 2; ++tj)
#pragma unroll
      for (int v = 0; v < 8; ++v)
        D[(ti * 16 + half + v) * sd + tj * 16 + col] = acc.t[ti][tj][v];
}

// Store only the first 8 columns of a TI x 1 tile result (narrow N matrices)
template <int TI>
__device__ __forceinline__ void store_n8_lds(const Acc& acc, float* D, int sd,
                                             int lane) {
  const int half = (lane >> 4) * 8;
  const int col = lane & 15;
  if (col < 8) {
#pragma unroll
    for (int ti = 0; ti < TI; ++ti)
#pragma unroll
      for (int v = 0; v < 8; ++v)
        D[(ti * 16 + half + v) * sd + col] = acc.t[ti][0][v];
  }
}

__global__ __launch_bounds__(32) void kalman_step_kernel(
    const float* __restrict__ gIn, const float* __restrict__ gMean,
    const float* __restrict__ gCov, const float* __restrict__ gH,
    const float* __restrict__ gR, const float* __restrict__ gF,
    const float* __restrict__ gQ, float* __restrict__ outMean,
    float* __restrict__ outCov) {
  __shared__ float sCov[32 * BIGS];    // cov, later new_cov
  __shared__ float sT[32 * BIGS];      // matmul temporary
  __shared__ float sA[32 * BIGS];      // ikh
  __shared__ float sF[32 * BIGS];      // F (async-staged early, used late)
  __shared__ float sH[16 * BIGS];      // H [8][32], rows 8..15 zero-padded
  __shared__ float sK[32 * 9];         // Kalman gain [32][8]
  __shared__ float sPHt[32 * 17];      // cov*H^T, later K*R (stride 17 as B)
  __shared__ float sR[8 * 9 + 24];     // R [8][8] (+pad: read as 16-wide B tile)
  __shared__ float sAug[8 * 17 + 24];  // [S | I] workspace (+pad for B reads)
  __shared__ float sMean[32];
  __shared__ float sMean2[32];
  __shared__ float sResid[8];
  __shared__ float sIn[8];

  const int lane = threadIdx.x;
  const size_t b = blockIdx.x;

  // ---- issue async staging: cov(8 chunks) + H(2) first, F(8) last ----------
  stage_mat<8>(gCov + b * 1024, sCov, lane);
  stage_mat<2>(gH + b * 256, sH, lane);
  stage_mat<8>(gF + b * 1024, sF, lane);  // completes while update phase runs

  // Q is consumed last via direct loads: warm L2 (global_prefetch_b8)
  __builtin_prefetch(gQ + b * 1024 + lane * 32, 0, 1);

  // small operands staged synchronously
  for (int i = lane; i < 8 * BIGS; i += 32) sH[8 * BIGS + i] = 0.f;  // zero pad
  for (int e = lane; e < 64; e += 32)
    sR[(e >> 3) * 9 + (e & 7)] = gR[b * 64 + e];
  sMean[lane] = gMean[b * 32 + lane];
  if (lane < 8) sIn[lane] = gIn[b * 8 + lane];

  wait_async<8>();  // cov + H complete (async loads finish in order); F pending
  __syncthreads();

  // ---------------- PHt = cov @ H^T  [32][8]  (WMMA, B = H^T via addressing) -
  {
    Acc a; acc_zero(a);
    mm_tiles<true, 32, 2, 1>(sCov, BIGS, sH, BIGS, lane, a);
    store_n8_lds<2>(a, sPHt, 17, lane);
  }
  __syncthreads();

  // ---------------- S = R + H @ PHt  [8][8]  (WMMA, R as C accumulator) ------
  {
    Acc a; acc_zero(a);
    const int col = lane & 15;
    if (lane < 16 && col < 8) {
#pragma unroll
      for (int v = 0; v < 8; ++v) a.t[0][0][v] = sR[v * 9 + col];
    }
    mm_tiles<false, 32, 1, 1>(sH, BIGS, sPHt, 17, lane, a);
    if (lane < 16 && col < 8) {
#pragma unroll
      for (int v = 0; v < 8; ++v) sAug[v * 17 + col] = a.t[0][0][v];
    }
    for (int e = lane; e < 64; e += 32) {
      int i = e >> 3, j = e & 7;
      sAug[i * 17 + 8 + j] = (i == j) ? 1.f : 0.f;
    }
  }
  __syncthreads();

  // ---------------- invert S (8x8 SPD, Gauss-Jordan, no pivoting) ------------
#pragma unroll
  for (int p = 0; p < 8; ++p) {
    float invp = 1.f / sAug[p * 17 + p];
    if (lane < 16) sAug[p * 17 + lane] *= invp;
    __syncthreads();
    int i = lane & 7;
    float f = sAug[i * 17 + p];  // read factor before col p is updated
    int cbase = (lane >> 3) * 4; // 4 lanes x 4 cols cover each row
    if (i != p) {
#pragma unroll
      for (int cc = 0; cc < 4; ++cc) {
        int c = cbase + cc;
        sAug[i * 17 + c] -= f * sAug[p * 17 + c];
        sAug[i * 17 + 8 + c] -= f * sAug[p * 17 + 8 + c];
      }
    }
    __syncthreads();
  }

  // ---------------- K = PHt @ Sinv  [32][8]  (WMMA, KD=8) --------------------
  {
    Acc a; acc_zero(a);
    mm_tiles<false, 8, 2, 1>(sPHt, 17, sAug + 8, 17, lane, a);
    store_n8_lds<2>(a, sK, 9, lane);
  }
  __syncthreads();

  // ---------------- mean update ----------------
  if (lane < 8) {
    float r = sIn[lane];
#pragma unroll
    for (int k = 0; k < 32; ++k) r -= sH[lane * BIGS + k] * sMean[k];
    sResid[lane] = r;
  }
  __syncthreads();
  {
    float nm = sMean[lane];
#pragma unroll
    for (int j = 0; j < 8; ++j) nm += sK[lane * 9 + j] * sResid[j];
    sMean2[lane] = nm;
  }
  __syncthreads();

  // ---------------- ikh = I - K @ H  (WMMA, KD=8) ----------------
  {
    Acc a; acc_zero(a);
    mm_tiles<false, 8, 2, 2>(sK, 9, sH, BIGS, lane, a);
    const int half = (lane >> 4) * 8;
    const int col = lane & 15;
#pragma unroll
    for (int ti = 0; ti < 2; ++ti)
#pragma unroll
      for (int tj = 0; tj < 2; ++tj)
#pragma unroll
        for (int v = 0; v < 8; ++v) {
          int r = ti * 16 + half + v, c = tj * 16 + col;
          sA[r * BIGS + c] = ((r == c) ? 1.f : 0.f) - a.t[ti][tj][v];
        }
  }
  __syncthreads();

  // ---------------- T = ikh @ cov ----------------
  {
    Acc a; acc_zero(a);
    mm_tiles<false, 32, 2, 2>(sA, BIGS, sCov, BIGS, lane, a);
    store32_lds(a, sT, BIGS, lane);
  }
  __syncthreads();

  // ---------------- KR = K @ R  [32][8]  (WMMA, KD=8; reuse sPHt) ------------
  {
    Acc a; acc_zero(a);
    mm_tiles<false, 8, 2, 1>(sK, 9, sR, 9, lane, a);
    store_n8_lds<2>(a, sPHt, 17, lane);
  }
  __syncthreads();

  // ---------------- new_cov = KR @ K^T + T @ ikh^T  (Joseph form) ------------
  {
    Acc a; acc_zero(a);
    mm_tiles<true, 8, 2, 2>(sPHt, 17, sK, 9, lane, a);     // K R K^T
    mm_tiles<true, 32, 2, 2>(sT, BIGS, sA, BIGS, lane, a); // + (ikh cov) ikh^T
    store32_lds(a, sCov, BIGS, lane);                      // cov dead; reuse
  }

  wait_async<0>();  // F staged (overlapped with the whole update phase)
  __syncthreads();

  // ---------------- predicted mean = F @ new_mean ----------------
  {
    float pm = 0.f;
#pragma unroll
    for (int j = 0; j < 32; ++j) pm += sF[lane * BIGS + j] * sMean2[j];
    outMean[b * 32 + lane] = pm;
  }

  // ---------------- T = F @ new_cov ----------------
  {
    Acc a; acc_zero(a);
    mm_tiles<false, 32, 2, 2>(sF, BIGS, sCov, BIGS, lane, a);
    store32_lds(a, sT, BIGS, lane);
  }
  __syncthreads();

  // ---------------- out_cov = T @ F^T + Q  (Q as C accumulator) --------------
  {
    Acc a;
    const int half = (lane >> 4) * 8;
    const int col = lane & 15;
#pragma unroll
    for (int ti = 0; ti < 2; ++ti)
#pragma unroll
      for (int tj = 0; tj < 2; ++tj)
#pragma unroll
        for (int v = 0; v < 8; ++v)
          a.t[ti][tj][v] =
              gQ[b * 1024 + (ti * 16 + half + v) * 32 + tj * 16 + col];
    mm_tiles<true, 32, 2, 2>(sT, BIGS, sF, BIGS, lane, a);
#pragma unroll
    for (int ti = 0; ti < 2; ++ti)
#pragma unroll
      for (int tj = 0; tj < 2; ++tj)
#pragma unroll
        for (int v = 0; v < 8; ++v)
          outCov[b * 1024 + (ti * 16 + half + v) * 32 + tj * 16 + col] =
              a.t[ti][tj][v];
  }
}

extern "C" void kernel_launch(void* const* d_in, const int* in_sizes, int n_in,
                              void* d_out, int out_size, void* d_ws,
                              size_t ws_size, hipStream_t stream) {
  const float* gIn = (const float*)d_in[0];    // [B,8]
  const float* gMean = (const float*)d_in[1];  // [B,32]
  const float* gCov = (const float*)d_in[2];   // [B,32,32]
  const float* gH = (const float*)d_in[3];     // [B,8,32]
  const float* gR = (const float*)d_in[4];     // [B,8,8]
  const float* gF = (const float*)d_in[5];     // [B,32,32]
  const float* gQ = (const float*)d_in[6];     // [B,32,32]

  const int B = in_sizes[0] / 8;
  float* outMean = (float*)d_out;
  float* outCov = outMean + (size_t)B * 32;

  hipLaunchKernelGGL(kalman_step_kernel, dim3(B), dim3(32), 0, stream, gIn,
                     gMean, gCov, gH, gR, gF, gQ, outMean, outCov);
}